// ImpulseEncoder_73967926772321
// MI455X (gfx1250) — compile-verified
//
#include <hip/hip_runtime.h>
#include <math.h>

// ImpulseEncoder for MI455X (gfx1250, wave32).
// V[dim] = tanh( (1-d) * sum_t d^(S-1-t) * E[idx_t][dim] ),  d = sigmoid(ssm_decay)
// Pass 1: per-chunk EMA partials (bandwidth-bound 64MB gather, coalesced float4).
// Pass 2: weighted chunk combine via chained V_WMMA_F32_16X16X4_F32 + tanh.

#define SEQ    8192
#define DIM    2048
#define NCHUNK 128
#define CLEN   (SEQ / NCHUNK)   // 64 tokens per chunk
#define KB     (NCHUNK / 4)     // 32 WMMA K-steps

typedef float v2f __attribute__((ext_vector_type(2)));
typedef float v8f __attribute__((ext_vector_type(8)));

__device__ __forceinline__ float sigmoidf_dev(float x) {
    return 1.0f / (1.0f + __expf(-x));
}

// ---------------------------------------------------------------------------
// Pass 1: one block per 64-token chunk; 512 threads x float4 cover DIM=2048.
// p_c = sum_{t in chunk} d^(chunk_end-1-t) * x_t   via   p = d*p + x_t
// ---------------------------------------------------------------------------
__global__ __launch_bounds__(512)
void ie_partial_kernel(const int* __restrict__ indices,
                       const float* __restrict__ embedding,
                       const float* __restrict__ ssm_decay,
                       float* __restrict__ P /* [NCHUNK][DIM] */) {
    __shared__ int sidx[CLEN];
    const int c   = blockIdx.x;
    const int tid = threadIdx.x;          // 0..511
    const float d = sigmoidf_dev(ssm_decay[0]);

    if (tid < CLEN) sidx[tid] = indices[c * CLEN + tid];
    __syncthreads();

    const int col = tid * 4;              // this thread's 4 dims
    float4 acc = make_float4(0.f, 0.f, 0.f, 0.f);

    for (int t = 0; t < CLEN; ++t) {
        const float* row = embedding + (size_t)sidx[t] * DIM + col;
        const int tn = (t + 1 < CLEN) ? t + 1 : t;
        // emits global_prefetch_b8 on gfx1250; hides gather latency
        __builtin_prefetch(embedding + (size_t)sidx[tn] * DIM + col, 0, 0);
        const float4 x = *(const float4*)row;
        acc.x = fmaf(d, acc.x, x.x);
        acc.y = fmaf(d, acc.y, x.y);
        acc.z = fmaf(d, acc.z, x.z);
        acc.w = fmaf(d, acc.w, x.w);
    }
    *(float4*)(P + (size_t)c * DIM + col) = acc;
}

// ---------------------------------------------------------------------------
// Pass 2: h[dim] = sum_c s_c * p_c[dim],  s_c = (1-d) * d^(CLEN*(NCHUNK-1-c))
// One wave per 16-dim tile. K=128 chunks consumed 4 at a time with chained
// V_WMMA_F32_16X16X4_F32 (C-accumulate). Scales are folded into the B rows
// and A == 1 everywhere, so the dot over K is invariant to the exact K-slot
// lane layout (each of the 4 chunks occupies exactly one K slot).
//
// Scales are generated iteratively (geometric sequence), walking kb DOWNWARD
// so the running scale shrinks and cleanly underflows to 0 for old chunks —
// no powf in the loop.
// ---------------------------------------------------------------------------
__global__ __launch_bounds__(256)
void ie_combine_kernel(const float* __restrict__ P,
                       const float* __restrict__ ssm_decay,
                       float* __restrict__ out) {
    const int lane = threadIdx.x & 31;
    const int wave = threadIdx.x >> 5;
    const int tile = blockIdx.x * (blockDim.x >> 5) + wave;  // 0..127 (DIM/16)
    const int n    = lane & 15;
    const int half = lane >> 4;
    const int dim  = tile * 16 + n;

    const float d  = sigmoidf_dev(ssm_decay[0]);
    const float om = 1.0f - d;

    // d^64 by repeated squaring; then d^128, d^256
    float d2   = d * d;
    float d4   = d2 * d2;
    float d8   = d4 * d4;
    float d16  = d8 * d8;
    float d32  = d16 * d16;
    float d64  = d32 * d32;
    float d128 = d64 * d64;
    float d256 = d128 * d128;

    // At kb = KB-1: c1 = 4*kb + 2*half + 1, exponent e1 = 64*(2 - 2*half)
    float s1 = om * (half ? 1.0f : d128);

    v8f acc = {0.f, 0.f, 0.f, 0.f, 0.f, 0.f, 0.f, 0.f};
    v2f a;  a.x = 1.0f;  a.y = 1.0f;   // A[m][k] == 1 for all m,k

    for (int kb = KB - 1; kb >= 0; --kb) {
        const int c0 = kb * 4 + 2 * half;   // this lane-half's two K slots
        const int c1 = c0 + 1;
        const float s0 = s1 * d64;          // e0 = e1 + 64
        v2f b;
        b.x = s0 * P[(size_t)c0 * DIM + dim];
        b.y = s1 * P[(size_t)c1 * DIM + dim];
        acc = __builtin_amdgcn_wmma_f32_16x16x4_f32(
            /*neg_a=*/false, a, /*neg_b=*/false, b,
            /*c_mod=*/(short)0, acc, /*reuse_a=*/false, /*reuse_b=*/false);
        s1 *= d256;                         // step to previous chunk group
    }

    // D rows are identical (A rows identical): M=0 row lives in acc[0],
    // lanes 0..15, N = lane.
    if (lane < 16) out[tile * 16 + lane] = tanhf(acc[0]);
}

// ---------------------------------------------------------------------------
extern "C" void kernel_launch(void* const* d_in, const int* in_sizes, int n_in,
                              void* d_out, int out_size, void* d_ws, size_t ws_size,
                              hipStream_t stream) {
    const int*   indices   = (const int*)d_in[0];     // [SEQ]
    const float* embedding = (const float*)d_in[1];   // [VOCAB, DIM]
    const float* ssm_decay = (const float*)d_in[2];   // [1]
    float*       out       = (float*)d_out;           // [DIM]
    float*       P         = (float*)d_ws;            // [NCHUNK][DIM] = 1 MB

    ie_partial_kernel<<<NCHUNK, 512, 0, stream>>>(indices, embedding, ssm_decay, P);

    // DIM/16 = 128 tiles, 8 waves (256 threads) per block -> 16 blocks
    ie_combine_kernel<<<(DIM / 16) / 8, 256, 0, stream>>>(P, ssm_decay, out);
}